// EMRouting_86182813762002
// MI455X (gfx1250) — compile-verified
//
#include <hip/hip_runtime.h>
#include <math.h>

typedef __attribute__((ext_vector_type(2))) float v2f;
typedef __attribute__((ext_vector_type(8))) float v8f;

#define NI 288   // KH*KW*CI
#define NJ 32    // CO
#define NP 16    // 4x4 pose
#define NBHW 576 // B*H*W
#define MIN_VAR_F 0.0005f
#define PI_F 3.14159265358979f

static __device__ __forceinline__ v8f wmma4(v2f a, v2f b, v8f c) {
  // D = A(16x4 f32) * B(4x16 f32) + C(16x16 f32)
  return __builtin_amdgcn_wmma_f32_16x16x4_f32(
      /*neg_a=*/false, a, /*neg_b=*/false, b,
      /*c_mod=*/(short)0, c, /*reuse_a=*/false, /*reuse_b=*/false);
}

// ---------------------------------------------------------------------------
// m-step: for each j owned by this wave, compute mu, 1/sigma^2, a_j,
// inv_norm_j, c3_j = sum_p mu^2/sigma^2, via WMMA reduction over i.
// ---------------------------------------------------------------------------
static __device__ __forceinline__ void m_step(
    const float* __restrict__ vbase,
    const float* __restrict__ beta_a, const float* __restrict__ beta_u,
    float* sR, float* sAct, float* sMu, float* sW,
    float* sAj, float* sInvNorm, float* sC3,
    float inv_temp, int wave, int lane)
{
  const int mrow  = lane & 15;        // A-matrix M index (pose p here)
  const int khalf = (lane >> 4) << 1; // K offset for this half-wave (0 or 2)

  for (int jj = 0; jj < 4; ++jj) {
    const int j = wave * 4 + jj;

    // sum_R = sum_i R[i,j]*act[i]  (wave butterfly reduction)
    float sumR = 0.f;
    for (int i = lane; i < NI; i += 32) sumR += sR[i * NJ + j] * sAct[i];
    for (int off = 16; off > 0; off >>= 1) sumR += __shfl_xor(sumR, off, 32);

    // WMMA accumulate: c1[p] = sum_i Rw*v, c2[p] = sum_i Rw*v^2
    v8f c1 = {0.f,0.f,0.f,0.f,0.f,0.f,0.f,0.f};
    v8f c2 = {0.f,0.f,0.f,0.f,0.f,0.f,0.f,0.f};
    for (int i0 = 0; i0 < NI; i0 += 4) {
      const int ia = i0 + khalf;
      v2f b; // B[k,n] = Rw[i0+k, j], replicated over all N columns
      b.x = sR[ia * NJ + j] * sAct[ia];
      b.y = sR[(ia + 1) * NJ + j] * sAct[ia + 1];
      const float* vp = vbase + ((size_t)ia * NJ + j) * NP + mrow;
      v2f a;  a.x = vp[0];       a.y = vp[NJ * NP];
      c1 = wmma4(a, b, c1);
      v2f a2; a2.x = a.x * a.x;  a2.y = a.y * a.y;
      c2 = wmma4(a2, b, c2);
    }

    // Finalize. All columns of D are identical; lane L<16 holds rows 0..7,
    // lane L>=16 holds rows 8..15 -> each half computes its 8 p's.
    const float inv_sumR = 1.0f / sumR;
    const int pb = (lane >> 4) * 8;
    float costp = 0.f, logp = 0.f, c3p = 0.f;
    float muv_arr[8], w_arr[8];
#pragma unroll
    for (int r = 0; r < 8; ++r) {
      float s1  = c1[r] * inv_sumR;
      float muv = s1 + 1e-7f;
      float sig = c2[r] * inv_sumR - 2.f * muv * s1 + muv * muv + MIN_VAR_F;
      float w   = 1.0f / sig;
      muv_arr[r] = muv;  w_arr[r] = w;
      costp += 0.5f * __logf(sig);            // log(sqrt(sigma^2))
      logp  += __logf(2.f * PI_F * sig);      // for norm
      c3p   += muv * muv * w;
    }
    float cost = costp + __shfl_xor(costp, 16, 32);
    float lsum = logp  + __shfl_xor(logp, 16, 32);
    float c3   = c3p   + __shfl_xor(c3p, 16, 32);

    float total_cost = sumR * (16.f * beta_u[j] + cost);
    float x  = inv_temp * (beta_a[j] - total_cost);
    float aj = 1.0f / (1.0f + __expf(-x));

    if (mrow == 0) {
#pragma unroll
      for (int r = 0; r < 8; ++r) {
        sMu[j * NP + pb + r] = muv_arr[r];
        sW [j * NP + pb + r] = w_arr[r];
      }
    }
    if (lane == 0) {
      sAj[j]      = aj;
      sInvNorm[j] = __expf(-0.5f * lsum);     // 1/sqrt(prod 2*pi*sigma^2)
      sC3[j]      = c3;
    }
  }
}

// ---------------------------------------------------------------------------
// E-step: expo[i,j] = -0.5 * sum_p (v-mu)^2 / sigma^2 via WMMA over p,
// then per-ci softmax-style renormalization of R over (kh,kw,Co).
// sR is overwritten: expo -> val -> new R.
// ---------------------------------------------------------------------------
static __device__ __forceinline__ void e_step(
    const float* __restrict__ vbase,
    float* sR, float* sMu, float* sW,
    float* sAj, float* sInvNorm, float* sC3, float* sRed,
    int tid, int wave, int lane)
{
  const int mrow  = lane & 15;
  const int khalf = (lane >> 4) << 1;

  // --- expo via WMMA (18 i-tiles x 32 j = 576 jobs over 8 waves) ---
  for (int t = wave; t < 18 * NJ; t += 8) {
    const int j  = t & 31;
    const int i0 = (t >> 5) * 16;
    v8f q1 = {0.f,0.f,0.f,0.f,0.f,0.f,0.f,0.f};
    v8f q2 = {0.f,0.f,0.f,0.f,0.f,0.f,0.f,0.f};
    const float* vp = vbase + ((size_t)(i0 + mrow) * NJ + j) * NP;
#pragma unroll
    for (int cch = 0; cch < 4; ++cch) {
      const int p0 = cch * 4 + khalf;
      // vp is 64B-aligned (row stride 64B, base 256B) and p0 is even ->
      // vp+p0 is 8B-aligned: load both K values as one b64.
      v2f a = *(const v2f*)(vp + p0);
      float w0 = sW[j * NP + p0],  w1 = sW[j * NP + p0 + 1];
      float m0 = sMu[j * NP + p0], m1 = sMu[j * NP + p0 + 1];
      v2f a2; a2.x = a.x * a.x;    a2.y = a.y * a.y;
      v2f b1; b1.x = w0;           b1.y = w1;       // weights 1/sigma^2
      v2f b2; b2.x = m0 * w0;      b2.y = m1 * w1;  // mu/sigma^2
      q1 = wmma4(a2, b1, q1);  // sum_p v^2 * w
      q2 = wmma4(a,  b2, q2);  // sum_p v * mu * w
    }
    if (mrow == 0) {
      const int rb = (lane >> 4) * 8;
      const float c3 = sC3[j];
#pragma unroll
      for (int r = 0; r < 8; ++r) {
        sR[(i0 + rb + r) * NJ + j] = -0.5f * (q1[r] - 2.f * q2[r] + c3);
      }
    }
  }
  __syncthreads();

  // --- per-ci normalization over (kh*kw, j): 32 ci groups x 8 threads ---
  const int ci = tid & 31;
  const int s  = tid >> 5;
  const int j0 = s * 4;

  float lmax = -1e30f;
  for (int kk = 0; kk < 9; ++kk)
    for (int dj = 0; dj < 4; ++dj)
      lmax = fmaxf(lmax, sR[(kk * 32 + ci) * NJ + j0 + dj]);
  sRed[ci * 8 + s] = lmax;
  __syncthreads();
  float gmax = -1e30f;
  for (int q = 0; q < 8; ++q) gmax = fmaxf(gmax, sRed[ci * 8 + q]);
  __syncthreads();

  float lsum = 0.f;
  for (int kk = 0; kk < 9; ++kk)
    for (int dj = 0; dj < 4; ++dj) {
      const int j = j0 + dj;
      const int idx = (kk * 32 + ci) * NJ + j;
      float val = __expf(sR[idx] - gmax) * sAj[j] * sInvNorm[j];
      sR[idx] = val;
      lsum += val;
    }
  sRed[ci * 8 + s] = lsum;
  __syncthreads();
  float gsum = 0.f;
  for (int q = 0; q < 8; ++q) gsum += sRed[ci * 8 + q];
  const float inv = 1.0f / gsum;
  for (int kk = 0; kk < 9; ++kk)
    for (int dj = 0; dj < 4; ++dj)
      sR[(kk * 32 + ci) * NJ + j0 + dj] *= inv;
  __syncthreads();
}

// ---------------------------------------------------------------------------
__global__ __launch_bounds__(256) void EMRouting_gfx1250_kernel(
    const float* __restrict__ votes, const float* __restrict__ acts,
    const float* __restrict__ beta_a, const float* __restrict__ beta_u,
    float* __restrict__ out)
{
  __shared__ float sR[NI * NJ];      // R / expo / val (aliased), 36 KB
  __shared__ float sAct[NI];
  __shared__ float sMu[NJ * NP];
  __shared__ float sW [NJ * NP];
  __shared__ float sAj[NJ];
  __shared__ float sInvNorm[NJ];
  __shared__ float sC3[NJ];
  __shared__ float sRed[32 * 8];

  const int tid  = threadIdx.x;
  const int wave = tid >> 5;
  const int lane = tid & 31;
  const int bhw  = blockIdx.x;
  const float* vbase = votes + (size_t)bhw * NI * NJ * NP;

  // Warm the L2 with this block's votes slice (576 KB): one prefetch per
  // 64B row -> emits global_prefetch_b8.
  for (int off = tid; off < NI * NJ; off += 256)
    __builtin_prefetch(vbase + (size_t)off * NP, 0, 1);

  for (int i = tid; i < NI; i += 256) sAct[i] = acts[bhw * NI + i];
  for (int idx = tid; idx < NI * NJ; idx += 256) sR[idx] = 1.0f; // softmax(0) over size-1 axis
  __syncthreads();

  // inv_temp = 1 - 0.95^(i+1) for i = 0, 1 and final i = ITERATIONS(=3)
  const float it0 = 0.05f, it1 = 0.0975f, itF = 0.18549375f;

  m_step(vbase, beta_a, beta_u, sR, sAct, sMu, sW, sAj, sInvNorm, sC3, it0, wave, lane);
  __syncthreads();
  e_step(vbase, sR, sMu, sW, sAj, sInvNorm, sC3, sRed, tid, wave, lane);

  m_step(vbase, beta_a, beta_u, sR, sAct, sMu, sW, sAj, sInvNorm, sC3, it1, wave, lane);
  __syncthreads();
  e_step(vbase, sR, sMu, sW, sAj, sInvNorm, sC3, sRed, tid, wave, lane);

  m_step(vbase, beta_a, beta_u, sR, sAct, sMu, sW, sAj, sInvNorm, sC3, itF, wave, lane);
  __syncthreads();

  // outputs: poses [B,H,W,Co,4,4] then acts [B,H,W,Co], concatenated flat
  float* out_poses = out;
  float* out_acts  = out + (size_t)NBHW * NJ * NP;
  // 512 pose floats per block: vector (v2f) writeback, 8B-aligned.
  for (int idx = tid; idx < (NJ * NP) / 2; idx += 256)
    *(v2f*)(out_poses + (size_t)bhw * NJ * NP + idx * 2) = *(const v2f*)(sMu + idx * 2);
  for (int j = tid; j < NJ; j += 256)
    out_acts[bhw * NJ + j] = sAj[j];
}

extern "C" void kernel_launch(void* const* d_in, const int* in_sizes, int n_in,
                              void* d_out, int out_size, void* d_ws, size_t ws_size,
                              hipStream_t stream) {
  const float* votes  = (const float*)d_in[0];
  const float* acts   = (const float*)d_in[1];
  const float* beta_a = (const float*)d_in[2];
  const float* beta_u = (const float*)d_in[3];
  float* out = (float*)d_out;
  (void)in_sizes; (void)n_in; (void)out_size; (void)d_ws; (void)ws_size;
  EMRouting_gfx1250_kernel<<<dim3(NBHW), dim3(256), 0, stream>>>(
      votes, acts, beta_a, beta_u, out);
}